// DCT_ATT_GCN_8572754723480
// MI455X (gfx1250) — compile-verified
//
#include <hip/hip_runtime.h>
#include <math.h>

// ---------------------------------------------------------------------------
// DCT-attention GCN for MI455X (gfx1250, wave32).
// fp32, ~110 GFLOP, ~25MB HBM -> compute bound -> native fp32 WMMA
// (V_WMMA_F32_16X16X4_F32). One workgroup per batch, y resident in LDS
// (320KB/WGP) across all 12 stages. 512 threads = 16 waves = 4 waves/SIMD for
// hardware latency hiding + depth-2 software pipelining of all fragments.
// Conv K-loops: unroll 2; others unroll 3.
// ---------------------------------------------------------------------------

typedef float v2f __attribute__((ext_vector_type(2)));
typedef float v8f __attribute__((ext_vector_type(8)));

#define WMMA4(c, a, b) \
  (c) = __builtin_amdgcn_wmma_f32_16x16x4_f32(false, (a), false, (b), (short)0, (c), false, false)

#define B_    128
#define NST_  12
#define NTH_  512
#define PI_F  3.14159265358979323846f
#define APAD_ 19584   // 144*136 padded adjacency matrix

// ---------------- prep kernels -------------------------------------------

extern "C" __global__ void k_prep_dct(float* __restrict__ dctm) {
  int i = blockIdx.x * 256 + threadIdx.x;
  if (i < 34 * 34) {
    int k = i / 34, ii = i % 34;
    float w = (k == 0) ? sqrtf(1.0f / 34.0f) : sqrtf(2.0f / 34.0f);
    dctm[i] = w * cosf(PI_F * (ii + 0.5f) * (float)k / 34.0f);
  }
}

extern "C" __global__ void k_prep_w(const float* __restrict__ kW1,
                                    const float* __restrict__ kW2,
                                    float* __restrict__ kW1p,
                                    float* __restrict__ kW2t) {
  const int N1 = 256 * 6 * 136, N2 = 256 * 5 * 256;
  for (int i = blockIdx.x * blockDim.x + threadIdx.x; i < N1; i += gridDim.x * blockDim.x) {
    int h = i / (6 * 136), r = i % (6 * 136), w = r / 136, d = r % 136;
    kW1p[i] = (d < 135) ? kW1[h * 810 + d * 6 + w] : 0.0f;
  }
  for (int i = blockIdx.x * blockDim.x + threadIdx.x; i < N2; i += gridDim.x * blockDim.x) {
    int h = i / (5 * 256), r = i % (5 * 256), w = r / 256, hp = r % 256;
    kW2t[i] = kW2[(h * 256 + hp) * 5 + w];
  }
}

extern "C" __global__ void k_prep_bn(const float* __restrict__ g, const float* __restrict__ be,
                                     const float* __restrict__ m, const float* __restrict__ v,
                                     float* __restrict__ sc, float* __restrict__ sh) {
  const int N = 25 * 34560;
  for (int i = blockIdx.x * blockDim.x + threadIdx.x; i < N; i += gridDim.x * blockDim.x) {
    float s = g[i] * rsqrtf(v[i] + 1e-5f);
    sc[i] = s;
    sh[i] = be[i] - m[i] * s;
  }
}

extern "C" __global__ void k_prep_a(const float* __restrict__ gc1_att,
                                    const float* __restrict__ blk_att,
                                    const float* __restrict__ gc7_att,
                                    float* __restrict__ Apad) {
  const int N = 26 * APAD_;
  for (int i = blockIdx.x * blockDim.x + threadIdx.x; i < N; i += gridDim.x * blockDim.x) {
    int g = i / APAD_, r = i % APAD_, n = r / 136, m = r % 136;
    const float* src = (g == 0) ? gc1_att : ((g <= 24) ? blk_att + (g - 1) * 18225 : gc7_att);
    Apad[i] = (n < 135 && m < 135) ? src[n * 135 + m] : 0.0f;
  }
}

// ---------------- K1: convs + attention + x = [dct_in | dct_att] ----------

extern "C" __global__ void k_att_x(const float* __restrict__ poses,
                                   const float* __restrict__ qW1,
                                   const float* __restrict__ qW2,
                                   const float* __restrict__ kW1p,
                                   const float* __restrict__ kW2t,
                                   const float* __restrict__ dctm,
                                   float* __restrict__ xws) {  // (B,144,80)
  extern __shared__ float lds[];
  float* pl  = lds;                    // 120*137
  float* k1l = pl + 120 * 137;         // 256*104 (reused as wsum 34*136)
  float* kl  = k1l + 256 * 104;        // 256*96
  float* q1l = kl + 256 * 96;          // 256*8
  float* ql  = q1l + 256 * 8;          // 256
  float* sl  = ql + 256;               // 128
  float* dl  = sl + 128;               // 34*34

  const int b = blockIdx.x;
  const int tid = threadIdx.x;
  const int wave = tid >> 5, lane = tid & 31, half = lane >> 4, l16 = lane & 15;
  const int kh = 2 * half;

  for (int i = tid; i < 120 * 137; i += NTH_) {
    int t = i / 137, d = i % 137;
    pl[i] = (d < 135) ? poses[(b * 120 + t) * 135 + d] : 0.0f;
  }
  for (int i = tid; i < 256 * 104; i += NTH_) k1l[i] = 0.0f;
  for (int i = tid; i < 34 * 34; i += NTH_) dl[i] = dctm[i];
  __syncthreads();

  // conv1-q (VALU, tiny): 1280 outputs over 512 threads
  for (int i = tid; i < 1280; i += NTH_) {
    int h = i / 5, t = i % 5;
    const float* w = qW1 + h * 810;
    float acc = 0.0f;
#pragma unroll 1
    for (int d = 0; d < 135; ++d)
#pragma unroll
      for (int wd = 0; wd < 6; ++wd)
        acc += pl[(110 + t + wd) * 137 + d] * w[d * 6 + wd];
    q1l[h * 8 + t] = fmaxf(acc, 0.0f);
  }

  // conv1-k WMMA: one M-group per wave, 6 N-tiles share the A-frag, depth-2 PF
  {
    const int h0 = wave * 16;
    v8f acc[6];
#pragma unroll
    for (int t = 0; t < 6; ++t) acc[t] = (v8f){};
#pragma unroll 1
    for (int w = 0; w < 6; ++w) {
      const float* arow = kW1p + ((h0 + l16) * 6 + w) * 136 + kh;
      auto ldB = [&](int t, int kk) {
        v2f r; int ka = kk * 4 + kh, rw = (t * 16 + l16 + w) * 137;
        r.x = pl[rw + ka]; r.y = pl[rw + ka + 1]; return r;
      };
      v2f aA = *(const v2f*)(arow), aB = *(const v2f*)(arow + 4), aC;
      v2f bA[6], bB[6], bC[6];
#pragma unroll
      for (int t = 0; t < 6; ++t) { bA[t] = ldB(t, 0); bB[t] = ldB(t, 1); }
#pragma unroll 2
      for (int kk = 0; kk < 32; ++kk) {
        aC = *(const v2f*)(arow + (kk + 2) * 4);
#pragma unroll
        for (int t = 0; t < 6; ++t) bC[t] = ldB(t, kk + 2);
#pragma unroll
        for (int t = 0; t < 6; ++t) WMMA4(acc[t], aA, bA[t]);
        aA = aB; aB = aC;
#pragma unroll
        for (int t = 0; t < 6; ++t) { bA[t] = bB[t]; bB[t] = bC[t]; }
      }
#pragma unroll
      for (int t = 0; t < 6; ++t) WMMA4(acc[t], aA, bA[t]);
#pragma unroll
      for (int t = 0; t < 6; ++t) WMMA4(acc[t], aB, bB[t]);
    }
#pragma unroll
    for (int t = 0; t < 6; ++t)
#pragma unroll
      for (int j = 0; j < 8; ++j)
        k1l[(h0 + j + 8 * half) * 104 + t * 16 + l16] = fmaxf(acc[t][j], 0.0f);
  }
  __syncthreads();

  // conv2-q (VALU)
  if (tid < 256) {
    const int h = tid;
    const float* w2 = qW2 + h * 1280;
    float acc = 0.0f;
#pragma unroll 1
    for (int hp = 0; hp < 256; ++hp)
#pragma unroll
      for (int wd = 0; wd < 5; ++wd)
        acc += q1l[hp * 8 + wd] * w2[hp * 5 + wd];
    ql[h] = fmaxf(acc, 0.0f);
  }

  // conv2-k WMMA: one M-group per wave, depth-2 PF
  {
    const int h0 = wave * 16;
    v8f acc[6];
#pragma unroll
    for (int t = 0; t < 6; ++t) acc[t] = (v8f){};
#pragma unroll 1
    for (int w = 0; w < 5; ++w) {
      const float* arow = kW2t + ((h0 + l16) * 5 + w) * 256 + kh;
      auto ldB = [&](int t, int kk) {
        v2f r; int ka = kk * 4 + kh, cc = t * 16 + l16 + w;
        r.x = k1l[ka * 104 + cc]; r.y = k1l[(ka + 1) * 104 + cc]; return r;
      };
      v2f aA = *(const v2f*)(arow), aB = *(const v2f*)(arow + 4), aC;
      v2f bA[6], bB[6], bC[6];
#pragma unroll
      for (int t = 0; t < 6; ++t) { bA[t] = ldB(t, 0); bB[t] = ldB(t, 1); }
#pragma unroll 2
      for (int kk = 0; kk < 62; ++kk) {
        aC = *(const v2f*)(arow + (kk + 2) * 4);
#pragma unroll
        for (int t = 0; t < 6; ++t) bC[t] = ldB(t, kk + 2);
#pragma unroll
        for (int t = 0; t < 6; ++t) WMMA4(acc[t], aA, bA[t]);
        aA = aB; aB = aC;
#pragma unroll
        for (int t = 0; t < 6; ++t) { bA[t] = bB[t]; bB[t] = bC[t]; }
      }
#pragma unroll
      for (int t = 0; t < 6; ++t) WMMA4(acc[t], aA, bA[t]);
#pragma unroll
      for (int t = 0; t < 6; ++t) WMMA4(acc[t], aB, bB[t]);
    }
#pragma unroll
    for (int t = 0; t < 6; ++t)
#pragma unroll
      for (int j = 0; j < 8; ++j)
        kl[(h0 + j + 8 * half) * 96 + t * 16 + l16] = fmaxf(acc[t][j], 0.0f);
  }
  __syncthreads();

  // score + normalization
  if (tid < 87) {
    float s = 1e-15f;
#pragma unroll 4
    for (int h = 0; h < 256; ++h) s += ql[h] * kl[h * 96 + tid];
    sl[tid] = s;
  }
  __syncthreads();
  if (tid == 0) {
    float tot = 0.0f;
    for (int n = 0; n < 87; ++n) tot += sl[n];
    sl[100] = 1.0f / tot;
  }
  __syncthreads();
  if (tid < 87) sl[tid] *= sl[100];
  __syncthreads();

  // wsum[v][d] = sum_n att[n]*poses[n+v][d]
  float* wsum = k1l;
  for (int i = tid; i < 34 * 135; i += NTH_) {
    int v = i / 135, d = i % 135;
    float acc = 0.0f;
#pragma unroll 4
    for (int n = 0; n < 87; ++n) acc += sl[n] * pl[(n + v) * 137 + d];
    wsum[v * 136 + d] = acc;
  }
  __syncthreads();

  // x[d][f] : f<34 dct_in, f>=34 dct_att ; zero-padded (144x80)
  for (int i = tid; i < 144 * 80; i += NTH_) {
    int d = i / 80, f = i % 80;
    float val = 0.0f;
    if (d < 135 && f < 68) {
      float acc = 0.0f;
      if (f < 34) {
#pragma unroll 4
        for (int v = 0; v < 34; ++v) {
          int row = (v < 10) ? (110 + v) : 119;
          acc += dl[f * 34 + v] * pl[row * 137 + d];
        }
      } else {
#pragma unroll 4
        for (int v = 0; v < 34; ++v) acc += dl[(f - 34) * 34 + v] * wsum[v * 136 + d];
      }
      val = acc;
    }
    xws[(b * 144 + d) * 80 + f] = val;
  }
}

// ---------------- K2: y = tanh(bn0( A1 @ x @ W1 + b1 )) -------------------

extern "C" __global__ void k_gc1(const float* __restrict__ A1p, const float* __restrict__ W1,
                                 const float* __restrict__ b1,
                                 const float* __restrict__ bnsc, const float* __restrict__ bnsh,
                                 const float* __restrict__ xws, float* __restrict__ yws) {
  extern __shared__ float lds[];
  float* xl = lds;             // 144*81
  float* tl = xl + 144 * 81;   // 144*81
  const int b = blockIdx.x, tid = threadIdx.x;
  const int wave = tid >> 5, lane = tid & 31, half = lane >> 4, l16 = lane & 15;
  const int kh = 2 * half;

  for (int i = tid; i < 144 * 81; i += NTH_) {
    int d = i / 81, f = i % 81;
    xl[i] = (f < 80) ? xws[(b * 144 + d) * 80 + f] : 0.0f;
  }
  __syncthreads();

  // t = A1 @ x : 45 tiles over 16 waves, K=34 steps, depth-2 PF
#pragma unroll 1
  for (int id = wave; id < 45; id += 16) {
    const int m0 = (id / 5) * 16, n0 = (id % 5) * 16;
    const float* arow = A1p + (m0 + l16) * 136 + kh;
    auto ldB = [&](int kk) {
      v2f r; int ka = kk * 4 + kh;
      r.x = xl[ka * 81 + n0 + l16]; r.y = xl[(ka + 1) * 81 + n0 + l16]; return r;
    };
    v8f c = {};
    v2f aA = *(const v2f*)(arow), aB = *(const v2f*)(arow + 4), aC;
    v2f bA = ldB(0), bB = ldB(1), bC;
#pragma unroll 3
    for (int kk = 0; kk < 32; ++kk) {
      aC = *(const v2f*)(arow + (kk + 2) * 4);
      bC = ldB(kk + 2);
      WMMA4(c, aA, bA);
      aA = aB; aB = aC; bA = bB; bB = bC;
    }
    WMMA4(c, aA, bA); WMMA4(c, aB, bB);
#pragma unroll
    for (int j = 0; j < 8; ++j) tl[(m0 + j + 8 * half) * 81 + n0 + l16] = c[j];
  }
  __syncthreads();

  // y = tanh(bn0(t @ W1 + b1)) : one N-tile per wave per M-block, K=17 steps
#pragma unroll 1
  for (int mt = 0; mt < 9; ++mt) {
    const int m0 = mt * 16, o0 = wave * 16;
    const float* trow = tl + (m0 + l16) * 81;
    auto ldA = [&](int kk) { v2f r; int ka = kk * 4 + kh; r.x = trow[ka]; r.y = trow[ka + 1]; return r; };
    auto ldW = [&](int kk) {
      v2f r; int ka = kk * 4 + kh;
      r.x = W1[ka * 256 + o0 + l16]; r.y = W1[(ka + 1) * 256 + o0 + l16]; return r;
    };
    v8f c = {};
    v2f aA = ldA(0), aB = ldA(1), aC;
    v2f bA = ldW(0), bB = ldW(1), bC;
#pragma unroll 3
    for (int kk = 0; kk < 15; ++kk) {
      aC = ldA(kk + 2); bC = ldW(kk + 2);
      WMMA4(c, aA, bA);
      aA = aB; aB = aC; bA = bB; bB = bC;
    }
    WMMA4(c, aA, bA); WMMA4(c, aB, bB);
#pragma unroll
    for (int j = 0; j < 8; ++j) {
      int n = m0 + j + 8 * half;
      float v0 = 0.0f;
      if (n < 135) {
        int i0 = n * 256 + o0 + l16;
        v0 = tanhf((c[j] + b1[o0 + l16]) * bnsc[i0] + bnsh[i0]);
      }
      yws[(b * 144 + n) * 256 + o0 + l16] = v0;
    }
  }
}

// ---------------- K3: 12 residual GCN stages, y resident in LDS ------------

extern "C" __global__ void k_stages(const float* __restrict__ Ablk,
                                    const float* __restrict__ blk_W,
                                    const float* __restrict__ blk_b,
                                    const float* __restrict__ bnsc,
                                    const float* __restrict__ bnsh,
                                    float* __restrict__ yws) {
  extern __shared__ float lds[];
  float* yl = lds;              // 144*257
  float* ul = yl + 144 * 257;   // 144*257
  const int b = blockIdx.x, tid = threadIdx.x;
  const int wave = tid >> 5, lane = tid & 31, half = lane >> 4, l16 = lane & 15;
  const int kh = 2 * half;

  for (int i = tid; i < 144 * 257; i += NTH_) {
    int n = i / 257, o = i % 257;
    yl[i] = (o < 256) ? yws[(b * 144 + n) * 256 + o] : 0.0f;
  }
  __syncthreads();

#pragma unroll 1
  for (int s = 0; s < NST_; ++s) {
    const float* Aap = Ablk + (2 * s) * APAD_;
    const float* Abp = Aap + APAD_;
    const float* Wa = blk_W + (2 * s) * 65536;
    const float* Wb = Wa + 65536;
    const float* ba = blk_b + (2 * s) * 256;
    const float* bbv = ba + 256;
    const float* sca = bnsc + (1 + 2 * s) * 34560;
    const float* scb = sca + 34560;
    const float* sha = bnsh + (1 + 2 * s) * 34560;
    const float* shb = sha + 34560;

    if (s + 1 < NST_) __builtin_prefetch(blk_W + (2 * s + 2) * 65536 + tid * 16, 0, 1);

    // phase1: u = Aa @ y ; one N-tile per wave per M-block, depth-2 PF
#pragma unroll 1
    for (int mt = 0; mt < 9; ++mt) {
      const int m0 = mt * 16, n0 = wave * 16;
      const float* arow = Aap + (m0 + l16) * 136 + kh;
      auto ldY = [&](int kk) {
        v2f r; int ka = kk * 4 + kh;
        r.x = yl[ka * 257 + n0 + l16]; r.y = yl[(ka + 1) * 257 + n0 + l16]; return r;
      };
      v8f c = {};
      v2f aA = *(const v2f*)(arow), aB = *(const v2f*)(arow + 4), aC;
      v2f bA = ldY(0), bB = ldY(1), bC;
#pragma unroll 3
      for (int kk = 0; kk < 32; ++kk) {
        aC = *(const v2f*)(arow + (kk + 2) * 4);
        bC = ldY(kk + 2);
        WMMA4(c, aA, bA);
        aA = aB; aB = aC; bA = bB; bB = bC;
      }
      WMMA4(c, aA, bA); WMMA4(c, aB, bB);
#pragma unroll
      for (int j = 0; j < 8; ++j) ul[(m0 + j + 8 * half) * 257 + n0 + l16] = c[j];
    }
    __syncthreads();

    // phase2: u <- tanh(bn(u @ Wa + ba)) in place, row-block staged
#pragma unroll 1
    for (int mt = 0; mt < 9; ++mt) {
      const int m0 = mt * 16, o0 = wave * 16;
      const float* urow = ul + (m0 + l16) * 257;
      auto ldA = [&](int kk) { v2f r; int ka = kk * 4 + kh; r.x = urow[ka]; r.y = urow[ka + 1]; return r; };
      auto ldW = [&](int kk) {
        v2f r; int ka = kk * 4 + kh;
        r.x = Wa[ka * 256 + o0 + l16]; r.y = Wa[(ka + 1) * 256 + o0 + l16]; return r;
      };
      v8f c = {};
      v2f aA = ldA(0), aB = ldA(1), aC;
      v2f bA = ldW(0), bB = ldW(1), bC;
#pragma unroll 3
      for (int kk = 0; kk < 62; ++kk) {
        aC = ldA(kk + 2); bC = ldW(kk + 2);
        WMMA4(c, aA, bA);
        aA = aB; aB = aC; bA = bB; bB = bC;
      }
      WMMA4(c, aA, bA); WMMA4(c, aB, bB);
      __syncthreads();  // all reads of rows m0..m0+15 complete
#pragma unroll
      for (int j = 0; j < 8; ++j) {
        int n = m0 + j + 8 * half;
        float v0 = 0.0f;
        if (n < 135) {
          int i0 = n * 256 + o0 + l16;
          v0 = tanhf((c[j] + ba[o0 + l16]) * sca[i0] + sha[i0]);
        }
        ul[n * 257 + o0 + l16] = v0;
      }
      // no barrier: next row-block reads disjoint rows
    }
    __syncthreads();

    // phase3: u <- Ab @ u ; each wave OWNS one column block -> barrier-free
    {
      const int n0 = wave * 16;
      v8f acc[9];
#pragma unroll
      for (int mt = 0; mt < 9; ++mt) acc[mt] = (v8f){};
      auto ldU = [&](int kk) {
        v2f r; int ka = kk * 4 + kh;
        r.x = ul[ka * 257 + n0 + l16]; r.y = ul[(ka + 1) * 257 + n0 + l16]; return r;
      };
      auto ldA = [&](int mt, int kk) {
        return *(const v2f*)(Abp + (mt * 16 + l16) * 136 + kk * 4 + kh);
      };
      v2f bA = ldU(0), bB = ldU(1), bC;
      v2f aA[9], aB[9], aC[9];
#pragma unroll
      for (int mt = 0; mt < 9; ++mt) { aA[mt] = ldA(mt, 0); aB[mt] = ldA(mt, 1); }
#pragma unroll 1
      for (int kk = 0; kk < 32; ++kk) {
        bC = ldU(kk + 2);
#pragma unroll
        for (int mt = 0; mt < 9; ++mt) aC[mt] = ldA(mt, kk + 2);
#pragma unroll
        for (int mt = 0; mt < 9; ++mt) WMMA4(acc[mt], aA[mt], bA);
        bA = bB; bB = bC;
#pragma unroll
        for (int mt = 0; mt < 9; ++mt) { aA[mt] = aB[mt]; aB[mt] = aC[mt]; }
      }
#pragma unroll
      for (int mt = 0; mt < 9; ++mt) WMMA4(acc[mt], aA[mt], bA);
#pragma unroll
      for (int mt = 0; mt < 9; ++mt) WMMA4(acc[mt], aB[mt], bB);
#pragma unroll
      for (int mt = 0; mt < 9; ++mt)
#pragma unroll
        for (int j = 0; j < 8; ++j)
          ul[(mt * 16 + j + 8 * half) * 257 + n0 + l16] = acc[mt][j];
    }
    __syncthreads();

    // phase4: y += tanh(bn(u @ Wb + bb))
#pragma unroll 1
    for (int mt = 0; mt < 9; ++mt) {
      const int m0 = mt * 16, o0 = wave * 16;
      const float* urow = ul + (m0 + l16) * 257;
      auto ldA = [&](int kk) { v2f r; int ka = kk * 4 + kh; r.x = urow[ka]; r.y = urow[ka + 1]; return r; };
      auto ldW = [&](int kk) {
        v2f r; int ka = kk * 4 + kh;
        r.x = Wb[ka * 256 + o0 + l16]; r.y = Wb[(ka + 1) * 256 + o0 + l16]; return r;
      };
      v8f c = {};
      v2f aA = ldA(0), aB = ldA(1), aC;
      v2f bA = ldW(0), bB = ldW(1), bC;
#pragma unroll 3
      for (int kk = 0; kk < 62; ++kk) {
        aC = ldA(kk + 2); bC = ldW(kk + 2);
        WMMA4(c, aA, bA);
        aA = aB; aB = aC; bA = bB; bB = bC;
      }
      WMMA4(c, aA, bA); WMMA4(c, aB, bB);
#pragma unroll
      for (int j = 0; j < 8; ++j) {
        int n = m0 + j + 8 * half;
        if (n < 135) {
          int i0 = n * 256 + o0 + l16;
          yl[n * 257 + o0 + l16] += tanhf((c[j] + bbv[o0 + l16]) * scb[i0] + shb[i0]);
        }
      }
    }
    __syncthreads();
  }

  for (int i = tid; i < 144 * 256; i += NTH_) {
    int n = i / 256, o = i % 256;
    yws[(b * 144 + n) * 256 + o] = yl[n * 257 + o];
  }
}

// ---------------- K4: gc7 + residual x + inverse DCT -> output -------------

extern "C" __global__ void k_out(const float* __restrict__ A7p, const float* __restrict__ W7,
                                 const float* __restrict__ b7, const float* __restrict__ dctm,
                                 const float* __restrict__ xws, const float* __restrict__ yws,
                                 float* __restrict__ out) {
  extern __shared__ float lds[];
  float* yl = lds;               // 144*257
  float* ul = yl + 144 * 257;    // 144*257
  float* dcto = ul + 144 * 257;  // 135*35
  float* dl = dcto + 135 * 35;   // 34*34
  const int b = blockIdx.x, tid = threadIdx.x;
  const int wave = tid >> 5, lane = tid & 31, half = lane >> 4, l16 = lane & 15;
  const int kh = 2 * half;

  for (int i = tid; i < 144 * 257; i += NTH_) {
    int n = i / 257, o = i % 257;
    yl[i] = (o < 256) ? yws[(b * 144 + n) * 256 + o] : 0.0f;
  }
  for (int i = tid; i < 34 * 34; i += NTH_) dl[i] = dctm[i];
  __syncthreads();

  // u = A7 @ y
#pragma unroll 1
  for (int mt = 0; mt < 9; ++mt) {
    const int m0 = mt * 16, n0 = wave * 16;
    const float* arow = A7p + (m0 + l16) * 136 + kh;
    auto ldY = [&](int kk) {
      v2f r; int ka = kk * 4 + kh;
      r.x = yl[ka * 257 + n0 + l16]; r.y = yl[(ka + 1) * 257 + n0 + l16]; return r;
    };
    v8f c = {};
    v2f aA = *(const v2f*)(arow), aB = *(const v2f*)(arow + 4), aC;
    v2f bA = ldY(0), bB = ldY(1), bC;
#pragma unroll 3
    for (int kk = 0; kk < 32; ++kk) {
      aC = *(const v2f*)(arow + (kk + 2) * 4);
      bC = ldY(kk + 2);
      WMMA4(c, aA, bA);
      aA = aB; aB = aC; bA = bB; bB = bC;
    }
    WMMA4(c, aA, bA); WMMA4(c, aB, bB);
#pragma unroll
    for (int j = 0; j < 8; ++j) ul[(m0 + j + 8 * half) * 257 + n0 + l16] = c[j];
  }
  __syncthreads();

  // dct_out[:, :34] = u @ W7 + b7 + x  (27 tiles over 16 waves)
#pragma unroll 1
  for (int id = wave; id < 27; id += 16) {
    const int m0 = (id / 3) * 16, f0 = (id % 3) * 16;
    const float* urow = ul + (m0 + l16) * 257;
    auto ldA = [&](int kk) { v2f r; int ka = kk * 4 + kh; r.x = urow[ka]; r.y = urow[ka + 1]; return r; };
    auto ldW = [&](int kk) {
      v2f r; int ka = kk * 4 + kh;
      r.x = W7[ka * 68 + f0 + l16]; r.y = W7[(ka + 1) * 68 + f0 + l16]; return r;
    };
    v8f c = {};
    v2f aA = ldA(0), aB = ldA(1), aC;
    v2f bA = ldW(0), bB = ldW(1), bC;
#pragma unroll 3
    for (int kk = 0; kk < 62; ++kk) {
      aC = ldA(kk + 2); bC = ldW(kk + 2);
      WMMA4(c, aA, bA);
      aA = aB; aB = aC; bA = bB; bB = bC;
    }
    WMMA4(c, aA, bA); WMMA4(c, aB, bB);
#pragma unroll
    for (int j = 0; j < 8; ++j) {
      int n = m0 + j + 8 * half, f = f0 + l16;
      if (n < 135 && f < 34)
        dcto[n * 35 + f] = c[j] + b7[f] + xws[(b * 144 + n) * 80 + f];
    }
  }
  __syncthreads();

  // out[b,t-10,d] = sum_f idct[t][f]*dct_out[d][f] ; idct = dct^T
  for (int i = tid; i < 24 * 135; i += NTH_) {
    int t = 10 + i / 135, d = i % 135;
    float acc = 0.0f;
#pragma unroll 4
    for (int f = 0; f < 34; ++f) acc += dl[f * 34 + t] * dcto[d * 35 + f];
    out[(b * 24 + (t - 10)) * 135 + d] = acc;
  }
}

// ---------------- host launcher -------------------------------------------

extern "C" void kernel_launch(void* const* d_in, const int* in_sizes, int n_in,
                              void* d_out, int out_size, void* d_ws, size_t ws_size,
                              hipStream_t stream) {
  const float* poses   = (const float*)d_in[0];
  const float* qW1     = (const float*)d_in[1];
  const float* qW2     = (const float*)d_in[2];
  const float* kW1     = (const float*)d_in[3];
  const float* kW2     = (const float*)d_in[4];
  const float* gc1_att = (const float*)d_in[5];
  const float* gc1_W   = (const float*)d_in[6];
  const float* gc1_b   = (const float*)d_in[7];
  const float* blk_att = (const float*)d_in[8];
  const float* blk_W   = (const float*)d_in[9];
  const float* blk_b   = (const float*)d_in[10];
  const float* gc7_att = (const float*)d_in[11];
  const float* gc7_W   = (const float*)d_in[12];
  const float* gc7_b   = (const float*)d_in[13];
  const float* bn_g    = (const float*)d_in[14];
  const float* bn_be   = (const float*)d_in[15];
  const float* bn_m    = (const float*)d_in[16];
  const float* bn_v    = (const float*)d_in[17];

  float* ws = (float*)d_ws;
  float* dctm = ws;                  // 1156
  float* kW1p = dctm + 1156;         // 208896
  float* kW2t = kW1p + 208896;       // 327680
  float* bnsc = kW2t + 327680;       // 864000
  float* bnsh = bnsc + 864000;       // 864000
  float* Apad = bnsh + 864000;       // 26*19584 = 509184
  float* xws  = Apad + 509184;       // 128*144*80  = 1474560
  float* yws  = xws + 1474560;       // 128*144*256 = 4718592

  const size_t lds1 = (size_t)(120 * 137 + 256 * 104 + 256 * 96 + 256 * 8 + 256 + 128 + 1156) * 4;
  const size_t lds2 = (size_t)(144 * 81 * 2) * 4;
  const size_t lds3 = (size_t)(144 * 257 * 2) * 4;
  const size_t lds4 = (size_t)(144 * 257 * 2 + 135 * 35 + 1156) * 4;

  (void)hipFuncSetAttribute((const void*)k_att_x,  hipFuncAttributeMaxDynamicSharedMemorySize, (int)lds1);
  (void)hipFuncSetAttribute((const void*)k_gc1,    hipFuncAttributeMaxDynamicSharedMemorySize, (int)lds2);
  (void)hipFuncSetAttribute((const void*)k_stages, hipFuncAttributeMaxDynamicSharedMemorySize, (int)lds3);
  (void)hipFuncSetAttribute((const void*)k_out,    hipFuncAttributeMaxDynamicSharedMemorySize, (int)lds4);

  k_prep_dct<<<dim3(5), dim3(256), 0, stream>>>(dctm);
  k_prep_w<<<dim3(512), dim3(256), 0, stream>>>(kW1, kW2, kW1p, kW2t);
  k_prep_bn<<<dim3(512), dim3(256), 0, stream>>>(bn_g, bn_be, bn_m, bn_v, bnsc, bnsh);
  k_prep_a<<<dim3(512), dim3(256), 0, stream>>>(gc1_att, blk_att, gc7_att, Apad);

  k_att_x<<<dim3(B_), dim3(NTH_), lds1, stream>>>(poses, qW1, qW2, kW1p, kW2t, dctm, xws);
  k_gc1<<<dim3(B_), dim3(NTH_), lds2, stream>>>(Apad, gc1_W, gc1_b, bnsc, bnsh, xws, yws);
  k_stages<<<dim3(B_), dim3(NTH_), lds3, stream>>>(Apad + APAD_, blk_W, blk_b, bnsc, bnsh, yws);
  k_out<<<dim3(B_), dim3(NTH_), lds4, stream>>>(Apad + 25 * APAD_, gc7_W, gc7_b, dctm, xws, yws,
                                                (float*)d_out);
}